// Matcher_45294725103772
// MI455X (gfx1250) — compile-verified
//
#include <hip/hip_runtime.h>
#include <hip/hip_bf16.h>

// Batched 256x256 Jonker-Volgenant LAP (Hungarian matcher), one workgroup
// per batch. CDNA5-specific: whole cost tile resident in the 320KB WGP LDS
// (256x257 f32 = 263KB), staged with global_load_async_to_lds_b32 (transpose
// scatter, bank-conflict-free via 257-stride), completion via s_wait_asynccnt.

#define LAP_N  256
#define LAP_NP 257   // padded row length: 257 % 64 == 1 -> conflict-free scatter
#define LAP_BIG 1.0e30f

struct LapSmem {
    float P[LAP_N * LAP_NP];   // transposed, sanitized, masked cost matrix
    float u[LAP_N];
    float v[LAP_N];
    float shortest[LAP_N];
    int   pathrow[LAP_N];      // reused as 'order' buffer at the end
    int   col4row[LAP_N];
    int   row4col[LAP_N];
    int   SC[LAP_N];
    int   SR[LAP_N];
    float rval[8];
    int   ridx[8];
    int   b_i;
    int   b_sink;
    int   nk;
    float b_min;
};

__global__ __launch_bounds__(LAP_N) void
Matcher_45294725103772_kernel(const float* __restrict__ costs,
                              const int* __restrict__ pad_mask,
                              float* __restrict__ out) {
    __shared__ LapSmem sm;
    const int tid  = threadIdx.x;
    const int k    = blockIdx.x;
    const int lane = tid & 31;
    const int wave = tid >> 5;

    // ---- n_k = popcount(pad_mask[k, :]) -------------------------------
    if (tid == 0) sm.nk = 0;
    __syncthreads();
    {
        int m = (pad_mask[k * LAP_N + tid] != 0) ? 1 : 0;
        unsigned long long bal = __ballot(m);
        if (lane == 0) atomicAdd(&sm.nk, (int)__popcll(bal));
    }

    // ---- async transposed copy: global -> LDS -------------------------
    // Element (p = pred, o = obj) of costs[k] goes to P[o*257 + p].
    // Lane `tid` takes o = tid: global reads coalesced across lanes,
    // LDS scatter stride 257 words -> conflict-free.
    {
        unsigned long long gbase64 =
            (unsigned long long)(costs + (size_t)k * LAP_N * LAP_N);
        unsigned ldsP  = (unsigned)(size_t)(void*)&sm.P[0];
        unsigned lbase = ldsP + (unsigned)tid * (LAP_NP * 4u);
        unsigned gofs  = (unsigned)tid * 4u;
#pragma unroll 4
        for (int p = 0; p < LAP_N; ++p) {
            unsigned la = lbase + (unsigned)p * 4u;
            unsigned ga = gofs + (unsigned)p * (LAP_N * 4u);
            asm volatile("global_load_async_to_lds_b32 %0, %1, %2"
                         :: "v"(la), "v"(ga), "s"(gbase64)
                         : "memory");
        }
        asm volatile("s_wait_asynccnt 0" ::: "memory");
    }
    __syncthreads();
    const int nk = sm.nk;

    // ---- in-place fixup of own row: nan_to_num + padded-row zeroing ---
    {
        float* row = &sm.P[tid * LAP_NP];
        if (tid >= nk) {
#pragma unroll 8
            for (int p = 0; p < LAP_N; ++p) row[p] = 0.0f;
        } else {
#pragma unroll 8
            for (int p = 0; p < LAP_N; ++p) {
                float f = row[p];
                unsigned b = __float_as_uint(f);
                if (((b >> 23) & 0xffu) == 0xffu) f = 1.0e6f; // nan / +-inf
                row[p] = f;
            }
        }
    }

    // ---- solver state init --------------------------------------------
    sm.u[tid] = 0.0f;
    sm.v[tid] = 0.0f;
    sm.col4row[tid] = -1;
    sm.row4col[tid] = -1;
    __syncthreads();

    // ---- JV shortest augmenting path, one augmentation per row --------
    for (int cur = 0; cur < LAP_N; ++cur) {
        sm.shortest[tid] = LAP_BIG;
        sm.pathrow[tid]  = -1;
        sm.SC[tid] = 0;
        sm.SR[tid] = 0;
        if (tid == 0) { sm.b_i = cur; sm.b_min = 0.0f; sm.b_sink = -1; }
        __syncthreads();

        for (;;) {
            const int   i      = sm.b_i;
            const float minval = sm.b_min;
            if (tid == 0) sm.SR[i] = 1;

            // relax column tid via row i
            float sh = sm.shortest[tid];
            const int sc = sm.SC[tid];
            const float r = minval + sm.P[i * LAP_NP + tid] - sm.u[i] - sm.v[tid];
            if (!sc && r < sh) {
                sh = r;
                sm.shortest[tid] = r;
                sm.pathrow[tid]  = i;
            }

            // masked argmin over 256 columns, lowest-index tie-break
            float val  = sc ? LAP_BIG : sh;
            int   idx_ = tid;
#pragma unroll
            for (int off = 16; off > 0; off >>= 1) {
                float ov = __shfl_xor(val, off, 32);
                int   oi = __shfl_xor(idx_, off, 32);
                if (ov < val || (ov == val && oi < idx_)) { val = ov; idx_ = oi; }
            }
            if (lane == 0) { sm.rval[wave] = val; sm.ridx[wave] = idx_; }
            __syncthreads();

            if (tid == 0) {
                float bv = sm.rval[0];
                int   bj = sm.ridx[0];
#pragma unroll
                for (int w = 1; w < 8; ++w) {
                    float wv = sm.rval[w];
                    if (wv < bv) { bv = wv; bj = sm.ridx[w]; }
                }
                sm.SC[bj] = 1;
                sm.b_min  = bv;
                int nxt = sm.row4col[bj];
                if (nxt < 0) sm.b_sink = bj; else sm.b_i = nxt;
            }
            __syncthreads();
            if (sm.b_sink >= 0) break;
        }

        // dual updates (Crouse 2016 / scipy lsap), before augmentation
        const float minval = sm.b_min;
        const int   sink   = sm.b_sink;
        if (tid == cur) {
            sm.u[tid] += minval;
        } else if (sm.SR[tid]) {
            int c = sm.col4row[tid];
            c = c < 0 ? 0 : c;                       // clip (defensive)
            sm.u[tid] += minval - sm.shortest[c];
        }
        if (sm.SC[tid]) sm.v[tid] -= (minval - sm.shortest[tid]);
        __syncthreads();

        // augment alternating path back to cur (sequential, thread 0)
        if (tid == 0) {
            int j = sink;
            for (;;) {
                const int i2 = sm.pathrow[j];
                sm.row4col[j] = i2;
                const int jn  = sm.col4row[i2];
                sm.col4row[i2] = j;
                if (i2 == cur) break;
                j = jn;
            }
        }
        __syncthreads();
    }

    // ---- output with padded-row fill ----------------------------------
    // used[j] <=> row4col[j] < nk ; unused columns, ascending, fill the padded rows.
    if (tid == 0) {
        int cnt = 0;
        for (int j = 0; j < LAP_N; ++j)
            if (sm.row4col[j] >= nk) sm.pathrow[cnt++] = j;  // reuse pathrow
    }
    __syncthreads();
    const int res = (tid < nk) ? sm.col4row[tid] : sm.pathrow[tid - nk];
    out[k * LAP_N + tid] = (float)res;
}

extern "C" void kernel_launch(void* const* d_in, const int* in_sizes, int n_in,
                              void* d_out, int out_size, void* d_ws, size_t ws_size,
                              hipStream_t stream) {
    (void)n_in; (void)d_ws; (void)ws_size; (void)out_size;
    const float* costs = (const float*)d_in[0];
    const int*   pad   = (const int*)d_in[1];
    float*       out   = (float*)d_out;
    const int B = in_sizes[1] / LAP_N;   // pad_mask is [B, 256]
    Matcher_45294725103772_kernel<<<B, LAP_N, 0, stream>>>(costs, pad, out);
}